// Net_8546984919135
// MI455X (gfx1250) — compile-verified
//
#include <hip/hip_runtime.h>

// ---------------------------------------------------------------------------
// Graph-attention aggregation for MI455X (gfx1250).
//   K1: fused node projection GEMM via V_WMMA_F32_16X16X4_F32, operands
//       staged through LDS; D-tile funneled back through LDS so the global
//       epilogue is one coalesced b128 store per thread.
//       Y[n, 0:8] = [a_dst(2), p(2), a_src(2), v(2)]  (p = pos @ W_pos^T)
//   K2: per-edge alpha + atomic segment-max (monotonic u32 key)
//   K3: per-edge exp / numerator+denominator atomic adds (fused softmax)
//   K4: per-node normalize  out = num / (denom + 1e-16)
// ---------------------------------------------------------------------------

typedef __attribute__((ext_vector_type(2))) float v2f;
typedef __attribute__((ext_vector_type(8))) float v8f;

__device__ __forceinline__ unsigned f2key(float f) {
  unsigned b = __float_as_uint(f);
  return (b & 0x80000000u) ? ~b : (b | 0x80000000u);  // order-preserving map
}
__device__ __forceinline__ float key2f(unsigned k) {
  unsigned b = (k & 0x80000000u) ? (k ^ 0x80000000u) : ~k;
  return __uint_as_float(b);
}

// Packed weight matrix Wc (8 x 15), row n, col k:
//   rows 0-1 : W_dst      rows 2-3 : W_pos (cols 0-3, else 0)
//   rows 4-5 : W_src      rows 6-7 : W_lin
__device__ __forceinline__ float wsel(int n, int k,
                                      const float* __restrict__ Wlin,
                                      const float* __restrict__ Wsrc,
                                      const float* __restrict__ Wdst,
                                      const float* __restrict__ Wpos) {
  if (k >= 15 || n >= 8) return 0.0f;
  if (n < 2) return Wdst[n * 15 + k];
  if (n < 4) return (k < 4) ? Wpos[(n - 2) * 4 + k] : 0.0f;
  if (n < 6) return Wsrc[(n - 4) * 15 + k];
  return Wlin[(n - 6) * 15 + k];
}

// ---------------------------------------------------------------------------
// K1: node projections. Block = 256 threads = 8 waves = 8 16-node tiles.
// LDS: lA[128][16] padded activations (reused as lOut[128][8] after compute),
//      lB[16][16] = Wc^T zero-padded.
// Each wave runs four chained V_WMMA_F32_16X16X4_F32 (K = 0,4,8,12).
// ISA VGPR layouts (CDNA5 ISA):
//   A 16x4 f32 : lanes 0-15 M=0..15 {VGPR0:K=0, VGPR1:K=1},
//                lanes 16-31        {VGPR0:K=2, VGPR1:K=3}
//   B 4x16 f32 : symmetric (lane = N, half selects K pair)
//   D 16x16 f32: VGPR r -> M = r (+8 for upper half), N = lane&15
// ---------------------------------------------------------------------------
__global__ __launch_bounds__(256) void node_proj_wmma(
    const float* __restrict__ x,
    const float* __restrict__ Wlin, const float* __restrict__ Wsrc,
    const float* __restrict__ Wdst, const float* __restrict__ Wpos,
    float* __restrict__ Y, int n_tiles, int n_nodes) {
  __shared__ float lA[128 * 16];  // [node_local][k]; later reused as lOut
  __shared__ float lB[16 * 16];   // [k][n] = Wc[n][k], zero padded

  const int tid       = (int)threadIdx.x;
  const int node_base = (int)blockIdx.x * 128;

  // --- build B tile once per block (branchy select runs once, not per step)
  {
    const int k = tid >> 4;
    const int n = tid & 15;
    lB[tid] = wsel(n, k, Wlin, Wsrc, Wdst, Wpos);
  }

  // --- stage 128 node rows, zero-padded, single predicated coalesced pass
#pragma unroll
  for (int t = tid; t < 128 * 16; t += 256) {
    const int nl   = t >> 4;
    const int k    = t & 15;
    const int node = node_base + nl;
    float v = 0.0f;
    if (k < 15 && node < n_nodes) v = x[(size_t)node * 15 + k];
    lA[t] = v;
  }
  __syncthreads();

  // --- per-wave WMMA over one 16-node tile
  const int tile_local = tid >> 5;
  const int tile       = (int)blockIdx.x * 8 + tile_local;
  const bool active    = (tile < n_tiles);  // wave-uniform
  const int lane = tid & 31;
  const int half = lane >> 4;  // 0: K pair {0,1}; 1: K pair {2,3}
  const int l15  = lane & 15;

  v8f acc = {};
  if (active) {  // EXEC all-ones inside (wave-uniform branch)
    const float* aRow = &lA[(tile_local * 16 + l15) * 16];
#pragma unroll
    for (int kk = 0; kk < 16; kk += 4) {
      const int k0 = kk + half * 2;
      v2f a, b;
      a.x = aRow[k0];            // contiguous pair -> ds_load_b64
      a.y = aRow[k0 + 1];
      b.x = lB[k0 * 16 + l15];
      b.y = lB[(k0 + 1) * 16 + l15];
      acc = __builtin_amdgcn_wmma_f32_16x16x4_f32(
          false, a, false, b, (short)0, acc, false, false);
    }
  }
  __syncthreads();  // all lA reads consumed; safe to reuse as lOut[128][8]

  float* lOut = lA;
  if (active && l15 < 8) {
    const int rowbase = tile_local * 16 + half * 8;
#pragma unroll
    for (int r = 0; r < 8; ++r)
      lOut[(rowbase + r) * 8 + l15] = acc[r];
  }
  __syncthreads();

  // --- single coalesced b128 store per thread (2 threads per node row)
  const int node = node_base + (tid >> 1);
  if (node < n_nodes) {
    const float4 vv = *(const float4*)&lOut[tid * 4];
    *(float4*)(Y + (size_t)node_base * 8 + tid * 4) = vv;
  }
}

// ---------------------------------------------------------------------------
// K-init: zero accumulators (idempotent per launch; harness poisons buffers).
// ---------------------------------------------------------------------------
__global__ __launch_bounds__(256) void init_acc(float* __restrict__ out,
                                                float* __restrict__ denom,
                                                unsigned* __restrict__ mkey,
                                                int n2) {
  int t = blockIdx.x * blockDim.x + threadIdx.x;
  if (t < n2) {
    out[t]   = 0.0f;
    denom[t] = 0.0f;
    mkey[t]  = 0u;  // below every real float key (0 maps to NaN sentinel)
  }
}

// ---------------------------------------------------------------------------
// K2: alpha = (a_dst[i]+p[i]) - (a_src[j]+p[j]) + b ; spill alpha, atomic max.
// ---------------------------------------------------------------------------
__global__ __launch_bounds__(256) void edge_pass1(
    const int* __restrict__ ei, int E,
    const float* __restrict__ Y, const float* __restrict__ bpos,
    float2* __restrict__ alpha, unsigned* __restrict__ mkey) {
  int e = blockIdx.x * blockDim.x + threadIdx.x;
  if (e >= E) return;
  const int i = ei[e];
  const int j = ei[E + e];

  const float4 di = *(const float4*)(Y + (size_t)i * 8);      // a_dst, p_i
  const float2 pj = *(const float2*)(Y + (size_t)j * 8 + 2);  // p_j
  const float2 as = *(const float2*)(Y + (size_t)j * 8 + 4);  // a_src

  const float ax = di.x - as.x + (di.z - pj.x) + bpos[0];
  const float ay = di.y - as.y + (di.w - pj.y) + bpos[1];
  alpha[e] = make_float2(ax, ay);

  atomicMax(&mkey[2 * i + 0], f2key(ax));
  atomicMax(&mkey[2 * i + 1], f2key(ay));
}

// ---------------------------------------------------------------------------
// K3: e = exp(alpha - m[i]); denom[i] += e; out[i] += e * (v[j] + delta)
//     delta recomputed as alpha - a_dst[i] + a_src[j] (node tables are L2-hot)
// ---------------------------------------------------------------------------
__global__ __launch_bounds__(256) void edge_pass2(
    const int* __restrict__ ei, int E,
    const float* __restrict__ Y, const float2* __restrict__ alpha,
    const unsigned* __restrict__ mkey,
    float* __restrict__ denom, float* __restrict__ out) {
  int e = blockIdx.x * blockDim.x + threadIdx.x;
  if (e >= E) return;
  const int i = ei[e];
  const int j = ei[E + e];

  const float2 al = alpha[e];
  const float  mx = key2f(mkey[2 * i + 0]);
  const float  my = key2f(mkey[2 * i + 1]);
  const float2 ad = *(const float2*)(Y + (size_t)i * 8);      // a_dst
  const float4 sv = *(const float4*)(Y + (size_t)j * 8 + 4);  // a_src, v

  const float ex = __expf(al.x - mx);
  const float ey = __expf(al.y - my);
  const float dx = al.x - ad.x + sv.x;  // delta.x
  const float dy = al.y - ad.y + sv.y;  // delta.y

  atomicAdd(&denom[2 * i + 0], ex);
  atomicAdd(&denom[2 * i + 1], ey);
  atomicAdd(&out[2 * i + 0], ex * (sv.z + dx));
  atomicAdd(&out[2 * i + 1], ey * (sv.w + dy));
}

// ---------------------------------------------------------------------------
// K4: normalize
// ---------------------------------------------------------------------------
__global__ __launch_bounds__(256) void finalize(float* __restrict__ out,
                                                const float* __restrict__ denom,
                                                int n2) {
  int t = blockIdx.x * blockDim.x + threadIdx.x;
  if (t < n2) out[t] = out[t] / (denom[t] + 1e-16f);
}

// ---------------------------------------------------------------------------
extern "C" void kernel_launch(void* const* d_in, const int* in_sizes, int n_in,
                              void* d_out, int out_size, void* d_ws, size_t ws_size,
                              hipStream_t stream) {
  const float* x    = (const float*)d_in[0];  // (N,15)
  const int*   ei   = (const int*)d_in[1];    // (2,E) flat
  const float* Wlin = (const float*)d_in[2];  // (2,15)
  const float* Wsrc = (const float*)d_in[3];  // (2,15)
  const float* Wdst = (const float*)d_in[4];  // (2,15)
  const float* Wpos = (const float*)d_in[5];  // (2,4)
  const float* bpos = (const float*)d_in[6];  // (2,)
  float* out = (float*)d_out;                 // (N,2)

  const int N  = in_sizes[0] / 15;
  const int E  = in_sizes[1] / 2;
  const int n2 = N * 2;

  // Workspace carve-up (256B aligned): Y | alpha | mkey | denom
  auto align = [](size_t v) { return (v + 255) & ~(size_t)255; };
  char*  w     = (char*)d_ws;
  float* Y     = (float*)w;                 size_t o = align((size_t)N * 8 * 4);
  float2* alpha = (float2*)(w + o);         o += align((size_t)E * 2 * 4);
  unsigned* mkey = (unsigned*)(w + o);      o += align((size_t)n2 * 4);
  float* denom = (float*)(w + o);           // o += align((size_t)n2 * 4);

  const int n_tiles = (N + 15) / 16;

  init_acc<<<(n2 + 255) / 256, 256, 0, stream>>>(out, denom, mkey, n2);

  // 8 waves / block -> 8 tiles (128 nodes) / block
  node_proj_wmma<<<(n_tiles + 7) / 8, 256, 0, stream>>>(
      x, Wlin, Wsrc, Wdst, Wpos, Y, n_tiles, N);

  edge_pass1<<<(E + 255) / 256, 256, 0, stream>>>(ei, E, Y, bpos, alpha, mkey);
  edge_pass2<<<(E + 255) / 256, 256, 0, stream>>>(ei, E, Y, alpha, mkey, denom, out);
  finalize<<<(n2 + 255) / 256, 256, 0, stream>>>(out, denom, n2);
}